// PixelContrastLoss_85676007620784
// MI455X (gfx1250) — compile-verified
//
#include <hip/hip_runtime.h>
#include <math.h>

typedef float v2f __attribute__((ext_vector_type(2)));
typedef float v8f __attribute__((ext_vector_type(8)));

#define A_N   10
#define V_N   100
#define D_N   256
#define C_N   19
#define M_N   5000
#define ROWS  (A_N * V_N)        /* 1000  anchor rows  */
#define COLS  (C_N * M_N)        /* 95000 contrast cols */
#define REALC ((C_N - 1) * M_N)  /* 90000 real (non-zero) cols */
#define TEMP    0.07f
#define COS_EPS 1e-8f

/* workspace layout (floats) */
#define WS_NX       0
#define WS_NY       1000
#define WS_SUMALL   96000
#define WS_SUMMATCH 97000
#define WS_POS      98000

/* K=256 fp32 WMMA micro-kernel, register-blocked over RT row sub-tiles.
 * A fragment (sub-tile rt): LDS row rt*16 + (lane&15), f32 pair at k + 2*(lane>>4).
 * B fragment: global row for column n = lane&15, same k pair.
 * One B load feeds RT independent WMMA accumulation chains (hides XDL latency,
 * divides B traffic by RT). Full EXEC, branch-free; msk zeroes OOB columns.   */
template <int RT>
__device__ __forceinline__ void wmma_k256_rt(const float* __restrict__ asrc,
                                             const float* __restrict__ bsrc,
                                             float msk, bool edge, v8f acc[RT]) {
    if (!edge) {
        #pragma unroll
        for (int k = 0; k < 256; k += 4) {
            v2f bf = *(const v2f*)(bsrc + k);
            #pragma unroll
            for (int rt = 0; rt < RT; ++rt) {
                v2f af = *(const v2f*)(asrc + rt * 16 * 256 + k);
                acc[rt] = __builtin_amdgcn_wmma_f32_16x16x4_f32(false, af, false, bf,
                                                                (short)0, acc[rt],
                                                                false, false);
            }
        }
    } else {
        #pragma unroll
        for (int k = 0; k < 256; k += 4) {
            v2f bf = *(const v2f*)(bsrc + k);
            bf *= msk;                       /* 0/1 lane mask, no EXEC games */
            #pragma unroll
            for (int rt = 0; rt < RT; ++rt) {
                v2f af = *(const v2f*)(asrc + rt * 16 * 256 + k);
                acc[rt] = __builtin_amdgcn_wmma_f32_16x16x4_f32(false, af, false, bf,
                                                                (short)0, acc[rt],
                                                                false, false);
            }
        }
    }
}

/* ---------------- init: zero accumulators + output ---------------- */
__global__ void init_k(float* ws, float* out) {
    int i = blockIdx.x * 256 + threadIdx.x;
    if (i < 3000) ws[WS_SUMALL + i] = 0.0f;   // sumAll, sumMatch, posPartial
    if (i == 0)   out[0] = 0.0f;
}

/* ---------------- norms: nx[1000], ny[95000] ---------------- */
__global__ __launch_bounds__(128) void norms_k(const float* __restrict__ X,
                                               const float* __restrict__ Q,
                                               float* __restrict__ ws) {
    int wid  = blockIdx.x * 4 + (threadIdx.x >> 5);
    int lane = threadIdx.x & 31;
    if (wid >= ROWS + COLS) return;
    const float* src = nullptr;
    if (wid < ROWS) {
        int r = wid, a = r % A_N, v = r / A_N;
        src = X + (size_t)(a * V_N + v) * D_N;
    } else {
        int j = wid - ROWS;
        if (j < REALC) src = Q + (size_t)(j + M_N) * D_N;  // contrast row j = queue row j+M
    }
    float s = 0.0f;
    if (src) {
        const float4* p = (const float4*)src + lane * 2;
        float4 u = p[0], w = p[1];
        s = u.x*u.x + u.y*u.y + u.z*u.z + u.w*u.w
          + w.x*w.x + w.y*w.y + w.z*w.z + w.w*w.w;
    }
    for (int o = 16; o >= 1; o >>= 1) s += __shfl_xor(s, o);
    if (lane == 0) {
        float nrm = sqrtf(s);
        if (wid < ROWS) ws[WS_NX + wid] = nrm;
        else {
            int j = wid - ROWS;
            ws[WS_NY + j] = (j < REALC) ? nrm : 0.0f;  // trailing zero rows: ny = 0
        }
    }
}

/* ---------------- pass1: full GEMM -> sum exp(all), sum exp(class-match) ----------------
 * grid = (16 row blocks of 64, 93 column chunks of 1024), block = 128 (4 waves).
 * Each wave: 16 column tiles of 16, 4 row sub-tiles, K = 256 fp32 WMMA.
 */
__global__ __launch_bounds__(128) void pass1_k(const float* __restrict__ X,
                                               const int*   __restrict__ y,
                                               const float* __restrict__ Q,
                                               float* __restrict__ ws) {
    __shared__ __align__(16) float aT[64 * 256];   /* 64 KB A panel */
    __shared__ float nxs[64];
    __shared__ int   ys[64];
    int tid = threadIdx.x, wv = tid >> 5, lane = tid & 31;
    int rowBase = blockIdx.x * 64;

    { /* stage 64x256 A panel (anchor row r lives at X[(r%A)*V + r/A]) */
        int m  = tid >> 1;            /* 0..63 */
        int k0 = (tid & 1) * 128;     /* 128 floats per thread */
        int r  = rowBase + m;
        const float* src = (r < ROWS)
            ? X + (size_t)((r % A_N) * V_N + (r / A_N)) * D_N + k0 : nullptr;
        float4* dst = (float4*)&aT[m * 256 + k0];
        for (int i = 0; i < 32; ++i)
            dst[i] = src ? ((const float4*)src)[i] : make_float4(0.f, 0.f, 0.f, 0.f);
        if (tid < 64) {
            int rr = rowBase + tid;
            nxs[tid] = (rr < ROWS) ? ws[WS_NX + rr] : 0.0f;
            ys[tid]  = (rr < ROWS) ? y[rr % A_N] : -1;
        }
    }
    __syncthreads();

    int half = lane >> 4, n = lane & 15, koff = half * 2;
    const float* asrc = &aT[n * 256 + koff];       /* sub-tile rt adds rt*16*256 */
    float accAll[4][8], accMat[4][8];
    #pragma unroll
    for (int rt = 0; rt < 4; ++rt)
        for (int g = 0; g < 8; ++g) { accAll[rt][g] = 0.f; accMat[rt][g] = 0.f; }
    int chunkCol = blockIdx.y * 1024;

    for (int t = wv; t < 64; t += 4) {
        int colBase = chunkCol + t * 16;
        int j = colBase + n;                               /* this lane's column   */
        bool edge = (colBase + 16 > REALC);                /* wave-uniform         */
        int   jc  = (j < REALC) ? j : 0;                   /* clamped (edge only)  */
        float msk = (j < REALC) ? 1.0f : 0.0f;
        const float* bsrc = Q + (size_t)(jc + M_N) * D_N + koff;
        __builtin_prefetch(bsrc + 64 * D_N, 0, 1);         /* wave's next tile     */

        v8f acc[4] = {{0.f,0.f,0.f,0.f,0.f,0.f,0.f,0.f},
                      {0.f,0.f,0.f,0.f,0.f,0.f,0.f,0.f},
                      {0.f,0.f,0.f,0.f,0.f,0.f,0.f,0.f},
                      {0.f,0.f,0.f,0.f,0.f,0.f,0.f,0.f}};
        wmma_k256_rt<4>(asrc, bsrc, msk, edge, acc);

        if (j < COLS) {
            float nyj = ws[WS_NY + j];
            int   cls = (j < REALC) ? (j / M_N + 1) : 0;
            #pragma unroll
            for (int rt = 0; rt < 4; ++rt) {
                for (int g = 0; g < 8; ++g) {
                    int mrow = rt * 16 + g + 8 * half;
                    if (rowBase + mrow < ROWS) {
                        float lg = acc[rt][g] / fmaxf(nxs[mrow] * nyj, COS_EPS) / TEMP;
                        float e  = expf(lg);
                        accAll[rt][g] += e;
                        if (cls == ys[mrow]) accMat[rt][g] += e;
                    }
                }
            }
        }
    }
    /* reduce across the 16 lanes of each half (C/D layout: N = lane&15) */
    #pragma unroll
    for (int rt = 0; rt < 4; ++rt) {
        for (int g = 0; g < 8; ++g) {
            float sa = accAll[rt][g], sm = accMat[rt][g];
            for (int o = 8; o >= 1; o >>= 1) { sa += __shfl_xor(sa, o); sm += __shfl_xor(sm, o); }
            if (n == 0) {
                int r = rowBase + rt * 16 + g + 8 * half;
                if (r < ROWS) {
                    atomicAdd(&ws[WS_SUMALL + r],   sa);
                    atomicAdd(&ws[WS_SUMMATCH + r], sm);
                }
            }
        }
    }
}

/* ---------------- pass2: positive class block per anchor -> posPartial[r] ----------------
 * grid = (4 v-blocks of 32, 20 column chunks of 256, 10 anchors). Rows of anchor a are
 * X[a][0..99] (contiguous); positive block for class c>=1 is queue rows [c*M, (c+1)*M).
 */
__global__ __launch_bounds__(128) void pass2_k(const float* __restrict__ X,
                                               const int*   __restrict__ y,
                                               const float* __restrict__ Q,
                                               float* __restrict__ ws) {
    int a = blockIdx.z;
    int c = y[a];
    if (c == 0) return;  // class-0 positives are the zero block; handled analytically
    __shared__ __align__(16) float aT[32 * 256];   /* 32 KB */
    __shared__ float nxs[32], negs[32];
    int tid = threadIdx.x, wv = tid >> 5, lane = tid & 31;
    int vBase = blockIdx.x * 32;

    {
        int m  = tid >> 2;            /* 0..31 */
        int k0 = (tid & 3) * 64;      /* 64 floats per thread */
        int v  = vBase + m;
        const float* src = (v < V_N) ? X + (size_t)(a * V_N + v) * D_N + k0 : nullptr;
        float4* dst = (float4*)&aT[m * 256 + k0];
        for (int i = 0; i < 16; ++i)
            dst[i] = src ? ((const float4*)src)[i] : make_float4(0.f, 0.f, 0.f, 0.f);
        if (tid < 32) {
            int v2 = vBase + tid;
            if (v2 < V_N) {
                int r = v2 * A_N + a;
                nxs[tid]  = ws[WS_NX + r];
                negs[tid] = ws[WS_SUMALL + r] - ws[WS_SUMMATCH + r];
            } else { nxs[tid] = 0.0f; negs[tid] = 0.0f; }
        }
    }
    __syncthreads();

    int half = lane >> 4, n = lane & 15, koff = half * 2;
    const float* asrc = &aT[n * 256 + koff];
    float accPos[2][8];
    #pragma unroll
    for (int rt = 0; rt < 2; ++rt)
        for (int g = 0; g < 8; ++g) accPos[rt][g] = 0.f;
    int chunk = blockIdx.y * 256;

    for (int t = wv; t < 16; t += 4) {
        int jb = chunk + t * 16;
        int jj = jb + n;                                   /* offset in class block */
        bool edge = (jb + 16 > M_N);                       /* wave-uniform          */
        int   jcl = (jj < M_N) ? jj : 0;
        float msk = (jj < M_N) ? 1.0f : 0.0f;
        const float* bsrc = Q + (size_t)(c * M_N + jcl) * D_N + koff;
        __builtin_prefetch(bsrc + 64 * D_N, 0, 1);

        v8f acc[2] = {{0.f,0.f,0.f,0.f,0.f,0.f,0.f,0.f},
                      {0.f,0.f,0.f,0.f,0.f,0.f,0.f,0.f}};
        wmma_k256_rt<2>(asrc, bsrc, msk, edge, acc);

        if (jj < M_N) {
            float nyj = ws[WS_NY + (c - 1) * M_N + jj];
            #pragma unroll
            for (int rt = 0; rt < 2; ++rt) {
                for (int g = 0; g < 8; ++g) {
                    int mrow = rt * 16 + g + 8 * half;
                    if (vBase + mrow < V_N) {
                        float lg = acc[rt][g] / fmaxf(nxs[mrow] * nyj, COS_EPS) / TEMP;
                        accPos[rt][g] += lg - logf(expf(lg) + negs[mrow]);
                    }
                }
            }
        }
    }
    #pragma unroll
    for (int rt = 0; rt < 2; ++rt) {
        for (int g = 0; g < 8; ++g) {
            float s = accPos[rt][g];
            for (int o = 8; o >= 1; o >>= 1) s += __shfl_xor(s, o);
            if (n == 0) {
                int v = vBase + rt * 16 + g + 8 * half;
                if (v < V_N) atomicAdd(&ws[WS_POS + v * A_N + a], s);
            }
        }
    }
}

/* ---------------- finalize: per-row loss, diagonal fix, mean ---------------- */
__global__ __launch_bounds__(128) void fin_k(const float* __restrict__ X,
                                             const int*   __restrict__ y,
                                             const float* __restrict__ Q,
                                             const float* __restrict__ ws,
                                             float* __restrict__ out) {
    int r    = blockIdx.x * 4 + (threadIdx.x >> 5);
    int lane = threadIdx.x & 31;
    if (r >= ROWS) return;
    int a = r % A_N, v = r / A_N;
    int c = y[a];
    float neg = ws[WS_SUMALL + r] - ws[WS_SUMMATCH + r];
    float pos, cnt;
    if (c == 0) {  // positives are zero columns: logit = 0, exp = 1
        pos = (float)M_N * (0.0f - logf(1.0f + neg));
        cnt = (float)M_N;
    } else {
        pos = ws[WS_POS + r];
        cnt = (float)M_N;
        if (c == 1) {  // logits_mask removes the [r,r] diagonal (column r is class 1)
            const float* ax = X + (size_t)(a * V_N + v) * D_N;
            const float* bx = Q + (size_t)(M_N + r) * D_N;
            float s = 0.0f;
            const float4* ap = (const float4*)ax + lane * 2;
            const float4* bp = (const float4*)bx + lane * 2;
            for (int i = 0; i < 2; ++i) {
                float4 u = ap[i], w = bp[i];
                s += u.x*w.x + u.y*w.y + u.z*w.z + u.w*w.w;
            }
            for (int o = 16; o >= 1; o >>= 1) s += __shfl_xor(s, o);
            float lg = s / fmaxf(ws[WS_NX + r] * ws[WS_NY + r], COS_EPS) / TEMP;
            pos -= lg - logf(expf(lg) + neg);
            cnt  = (float)(M_N - 1);
        }
    }
    if (lane == 0)
        atomicAdd(out, -(pos / (cnt + 1e-5f)) * (1.0f / (float)ROWS));
}

extern "C" void kernel_launch(void* const* d_in, const int* in_sizes, int n_in,
                              void* d_out, int out_size, void* d_ws, size_t ws_size,
                              hipStream_t stream) {
    const float* X  = (const float*)d_in[0];   // X_anchor [10,100,256] f32
    const int*   yA = (const int*)  d_in[1];   // y_anchor [10] i32
    const float* Q  = (const float*)d_in[2];   // queue    [19,5000,256] f32
    float* out = (float*)d_out;
    float* ws  = (float*)d_ws;

    init_k<<<12, 256, 0, stream>>>(ws, out);
    norms_k<<<(ROWS + COLS + 3) / 4, 128, 0, stream>>>(X, Q, ws);

    dim3 g1((ROWS + 63) / 64, (COLS + 1023) / 1024);   // 16 x 93
    pass1_k<<<g1, 128, 0, stream>>>(X, yA, Q, ws);

    dim3 g2((V_N + 31) / 32, (M_N + 255) / 256, A_N);  // 4 x 20 x 10
    pass2_k<<<g2, 128, 0, stream>>>(X, yA, Q, ws);

    fin_k<<<(ROWS + 3) / 4, 128, 0, stream>>>(X, yA, Q, ws, out);
}